// DistanceAwareGATv2_27118423507679
// MI455X (gfx1250) — compile-verified
//
#include <hip/hip_runtime.h>
#include <math.h>

#define NN 10000
#define EE 160000
#define KDIM 256
#define HH 4
#define CCH 64
#define OUTCH 256   // H*C

typedef __attribute__((ext_vector_type(2))) float v2f;
typedef __attribute__((ext_vector_type(8))) float v8f;

// Order-preserving float<->uint encoding for atomicMax on floats.
__device__ __forceinline__ unsigned enc_f32(float f) {
    unsigned u = __float_as_uint(f);
    return (u & 0x80000000u) ? ~u : (u | 0x80000000u);
}
__device__ __forceinline__ float dec_f32(unsigned u) {
    u = (u & 0x80000000u) ? (u & 0x7FFFFFFFu) : ~u;
    return __uint_as_float(u);
}

// ---------------------------------------------------------------------------
// Init: zero output + alpha_sum, reset global-max accumulator.
// enc(x) > 0 for every finite float, so 0 acts as -inf sentinel.
// ---------------------------------------------------------------------------
__global__ void __launch_bounds__(256) k_init(float* out, float* asum, unsigned* gmax) {
    size_t i = (size_t)blockIdx.x * 256 + threadIdx.x;
    size_t tot = (size_t)NN * OUTCH;
    for (size_t k = i; k < tot; k += (size_t)gridDim.x * 256) out[k] = 0.0f;
    if (i < (size_t)NN * HH) asum[i] = 0.0f;
    if (i == 0) *gmax = 0u;
}

// ---------------------------------------------------------------------------
// Fold the distance-MLP second layer into the attention vector:
//   G[j,h]  = sum_p de_w2[j,p] * attn[h, 2C+p]
//   ch[h]   = sum_p de_b2[p]   * attn[h, 2C+p]
// so per-edge pe[h] = ch[h] + sum_j relu(ed*w1[j]+b1[j]) * G[j,h]
// ---------------------------------------------------------------------------
__global__ void k_gtab(const float* __restrict__ de_w2, const float* __restrict__ de_b2,
                       const float* __restrict__ attn, float* __restrict__ g,
                       float* __restrict__ ch) {
    int t = threadIdx.x;
    if (t < 64) {
        int j = t >> 2, h = t & 3;
        float s = 0.0f;
        for (int p = 0; p < 32; ++p) s += de_w2[j * 32 + p] * attn[h * 160 + 128 + p];
        g[t] = s;  // g[j*4 + h]
    } else if (t < 68) {
        int h = t - 64;
        float s = 0.0f;
        for (int p = 0; p < 32; ++p) s += de_b2[p] * attn[h * 160 + 128 + p];
        ch[h] = s;
    }
}

// ---------------------------------------------------------------------------
// x_proj = x @ W_lin + b  via V_WMMA_F32_16X16X4_F32 (fp32, matches reference).
// Block: 256 thr = 8 waves; 16 output rows per block, wave w owns cols
// [32w, 32w+32) = two 16x16 WMMA tiles. K loop steps by 4.
// A frag (16x4): lanes 0-15 K={0,1}, lanes 16-31 K={2,3} -> one b64 load.
// B frag (4x16): same K-halving, N = lane&15.
// C/D: VGPR v -> row v + 8*(lane>=16), col = lane&15.
// ---------------------------------------------------------------------------
__global__ void __launch_bounds__(256) k_gemm(const float* __restrict__ x,
                                              const float* __restrict__ W,
                                              const float* __restrict__ bias,
                                              float* __restrict__ xproj) {
    const int lane  = threadIdx.x & 31;
    const int wave  = threadIdx.x >> 5;
    const int r0    = blockIdx.x * 16;
    const int arow  = lane & 15;
    const int khalf = (lane >> 4) << 1;        // 0 or 2
    const int col0  = wave * 32 + (lane & 15);
    const int col1  = col0 + 16;

    v8f c0 = {};
    v8f c1 = {};
    const float* aptr = x + (size_t)(r0 + arow) * KDIM + khalf;

    for (int k = 0; k < KDIM; k += 4) {
        v2f a = *(const v2f*)(aptr + k);
        const float* w0 = W + (size_t)(k + khalf) * OUTCH;
        v2f b0 = { w0[col0], w0[OUTCH + col0] };
        v2f b1v = { w0[col1], w0[OUTCH + col1] };
        c0 = __builtin_amdgcn_wmma_f32_16x16x4_f32(false, a, false, b0,  (short)0, c0, false, false);
        c1 = __builtin_amdgcn_wmma_f32_16x16x4_f32(false, a, false, b1v, (short)0, c1, false, false);
    }

    const int rowoff = (lane >> 4) << 3;
    const float bb0 = bias[col0];
    const float bb1 = bias[col1];
#pragma unroll
    for (int v = 0; v < 8; ++v) {
        int row = r0 + v + rowoff;
        xproj[(size_t)row * OUTCH + col0] = c0[v] + bb0;
        xproj[(size_t)row * OUTCH + col1] = c1[v] + bb1;
    }
}

// ---------------------------------------------------------------------------
// Per-node attention dots: a_src[n,h] = <x_proj[n,h,:], attn[h,:C]>,
//                          a_dst[n,h] = <x_proj[n,h,:], attn[h,C:2C]>
// One block per node; 256 threads = (h, c); LDS segmented tree-reduce.
// ---------------------------------------------------------------------------
__global__ void __launch_bounds__(256) k_nodedots(const float* __restrict__ xproj,
                                                  const float* __restrict__ attn,
                                                  float* __restrict__ a_src,
                                                  float* __restrict__ a_dst) {
    __shared__ float ss[256];
    __shared__ float sd[256];
    const int n = blockIdx.x;
    const int t = threadIdx.x;
    const int h = t >> 6, c = t & 63;
    float v = xproj[(size_t)n * OUTCH + t];
    ss[t] = v * attn[h * 160 + c];
    sd[t] = v * attn[h * 160 + 64 + c];
    __syncthreads();
    for (int off = 32; off; off >>= 1) {
        if (c < off) { ss[t] += ss[t + off]; sd[t] += sd[t + off]; }
        __syncthreads();
    }
    if (c == 0) {
        a_src[n * HH + h] = ss[t];
        a_dst[n * HH + h] = sd[t];
    }
}

// ---------------------------------------------------------------------------
// Per-edge pre-softmax logits (leaky-relu applied) + global max via atomicMax.
// ---------------------------------------------------------------------------
__global__ void __launch_bounds__(256) k_logits(const int* __restrict__ ei,
                                                const float* __restrict__ dist,
                                                const float* __restrict__ w1,
                                                const float* __restrict__ b1,
                                                const float* __restrict__ g,
                                                const float* __restrict__ ch,
                                                const float* __restrict__ a_src,
                                                const float* __restrict__ a_dst,
                                                float* __restrict__ logits,
                                                unsigned* __restrict__ gmax) {
    __shared__ float sm[256];
    const int e = blockIdx.x * 256 + threadIdx.x;
    float m = -3.4e38f;
    if (e < EE) {
        const int s = ei[e];
        const int d = ei[EE + e];
        const float ed = dist[(size_t)s * NN + d];
        float h1[16];
#pragma unroll
        for (int j = 0; j < 16; ++j) {
            float t = ed * w1[j] + b1[j];
            h1[j] = t > 0.0f ? t : 0.0f;
        }
#pragma unroll
        for (int h = 0; h < HH; ++h) {
            float pe = ch[h];
#pragma unroll
            for (int j = 0; j < 16; ++j) pe += h1[j] * g[j * 4 + h];
            float a = a_src[s * HH + h] + a_dst[d * HH + h] + pe;
            a = (a >= 0.0f) ? a : 0.2f * a;          // leaky_relu(0.2)
            logits[(size_t)e * HH + h] = a;
            m = fmaxf(m, a);
        }
    }
    sm[threadIdx.x] = m;
    __syncthreads();
    for (int off = 128; off; off >>= 1) {
        if (threadIdx.x < off) sm[threadIdx.x] = fmaxf(sm[threadIdx.x], sm[threadIdx.x + off]);
        __syncthreads();
    }
    if (threadIdx.x == 0) atomicMax(gmax, enc_f32(sm[0]));
}

// ---------------------------------------------------------------------------
// alpha = exp(logit - gmax); segment-sum into alpha_sum[dst,h].
// ---------------------------------------------------------------------------
__global__ void __launch_bounds__(256) k_alpha(const int* __restrict__ ei,
                                               float* __restrict__ logits,   // becomes alpha
                                               const unsigned* __restrict__ gmax,
                                               float* __restrict__ asum) {
    const int e = blockIdx.x * 256 + threadIdx.x;
    if (e >= EE) return;
    const float gm = dec_f32(*gmax);
    const int d = ei[EE + e];
#pragma unroll
    for (int h = 0; h < HH; ++h) {
        float a = __expf(logits[(size_t)e * HH + h] - gm);
        logits[(size_t)e * HH + h] = a;
        atomicAdd(asum + d * HH + h, a);
    }
}

// ---------------------------------------------------------------------------
// out[dst,h,c] += (alpha[e,h]/alpha_sum[dst,h]) * x_proj[src,h,c]
// One block per edge; 256 threads map (h,c). All traffic L2-resident.
// ---------------------------------------------------------------------------
__global__ void __launch_bounds__(256) k_scatter(const int* __restrict__ ei,
                                                 const float* __restrict__ xproj,
                                                 const float* __restrict__ alpha,
                                                 const float* __restrict__ asum,
                                                 float* __restrict__ out) {
    const int e = blockIdx.x;
    const int t = threadIdx.x;
    const int h = t >> 6;
    const int s = ei[e];
    const int d = ei[EE + e];
    const float w = alpha[(size_t)e * HH + h] / asum[d * HH + h];
    atomicAdd(out + (size_t)d * OUTCH + t, xproj[(size_t)s * OUTCH + t] * w);
}

extern "C" void kernel_launch(void* const* d_in, const int* in_sizes, int n_in,
                              void* d_out, int out_size, void* d_ws, size_t ws_size,
                              hipStream_t stream) {
    const float* x    = (const float*)d_in[0];
    const int*   ei   = (const int*)d_in[1];
    // d_in[2] = edge_attr (unused by reference math)
    const float* dist = (const float*)d_in[3];
    const float* W    = (const float*)d_in[4];
    const float* bl   = (const float*)d_in[5];
    const float* attn = (const float*)d_in[6];
    const float* w1   = (const float*)d_in[7];
    const float* b1   = (const float*)d_in[8];
    const float* w2   = (const float*)d_in[9];
    const float* b2   = (const float*)d_in[10];
    float* out = (float*)d_out;

    // Workspace layout (floats)
    float* ws     = (float*)d_ws;
    float* xproj  = ws;                                   // N*256  = 2,560,000
    float* a_src  = xproj + (size_t)NN * OUTCH;           // N*H    = 40,000
    float* a_dst  = a_src + (size_t)NN * HH;              // N*H
    float* logits = a_dst + (size_t)NN * HH;              // E*H    = 640,000 (reused as alpha)
    float* asum   = logits + (size_t)EE * HH;             // N*H
    unsigned* gmax = (unsigned*)(asum + (size_t)NN * HH); // 1
    float* gtab   = (float*)(gmax + 1);                   // 64
    float* ch     = gtab + 64;                            // 4

    k_init    <<<10000, 256, 0, stream>>>(out, asum, gmax);
    k_gtab    <<<1, 128, 0, stream>>>(w2, b2, attn, gtab, ch);
    k_gemm    <<<NN / 16, 256, 0, stream>>>(x, W, bl, xproj);       // 625 blocks
    k_nodedots<<<NN, 256, 0, stream>>>(xproj, attn, a_src, a_dst);
    k_logits  <<<EE / 256, 256, 0, stream>>>(ei, dist, w1, b1, gtab, ch,
                                             a_src, a_dst, logits, gmax);
    k_alpha   <<<EE / 256, 256, 0, stream>>>(ei, logits, gmax, asum);
    k_scatter <<<EE, 256, 0, stream>>>(ei, xproj, logits, asum, out);
}